// ControlledLayer_1262720385761
// MI455X (gfx1250) — compile-verified
//
#include <hip/hip_runtime.h>
#include <hip/hip_bf16.h>

#define FAN 8192
#define NSPLIT 16
#define KCHUNK (FAN / NSPLIT)   // 512

typedef __attribute__((ext_vector_type(2))) float v2f;
typedef __attribute__((ext_vector_type(4))) float v4f;
typedef __attribute__((ext_vector_type(8))) float v8f;

#if defined(__has_builtin)
#if __has_builtin(__builtin_amdgcn_wmma_f32_16x16x4_f32)
#define HAVE_WMMA_F32X4 1
#endif
#endif

// ---------------------------------------------------------------------------
// Kernel A: split-K mat-vec partials via V_WMMA_F32_16X16X4_F32.
// One wave owns a 16-row tile of W and a K-chunk. B operand broadcasts x
// across all 16 columns, so D column 0 == W_tile @ x. Lane layout (ISA 7.12.2):
//   A 16x4 f32: lanes 0-15 -> {K=0,K=1}, lanes 16-31 -> {K=2,K=3}, M = lane&15
//   D 16x16 f32: lane 0 holds M=0..7 (VGPR 0..7), lane 16 holds M=8..15
// Per 16-K step: 2 x b128 W loads per lane feed 4 WMMAs (K remapped so each
// lane's K values are contiguous: lanes<16 take k..k+7, lanes>=16 take k+8..k+15).
// ---------------------------------------------------------------------------
__global__ __launch_bounds__(256) void matvec_partial_kernel(
    const float* __restrict__ W, const float* __restrict__ x,
    float* __restrict__ partial)
{
    const int lane    = threadIdx.x & 31;
    const int wave    = threadIdx.x >> 5;
    const int rowTile = blockIdx.x * 8 + wave;   // 0..511
    const int split   = blockIdx.y;              // 0..NSPLIT-1
    const int rowBase = rowTile * 16;
    const int m       = lane & 15;
    const int half    = lane >> 4;               // 0: lanes 0-15, 1: lanes 16-31
    const int k0      = split * KCHUNK;

    v8f acc = {0.f, 0.f, 0.f, 0.f, 0.f, 0.f, 0.f, 0.f};

#ifdef HAVE_WMMA_F32X4
    const float* __restrict__ Wrow = W + (size_t)(rowBase + m) * FAN;
    for (int k = k0; k < k0 + KCHUNK; k += 16) {
        const int kk = k + half * 8;
        v4f wa = *(const v4f*)(Wrow + kk);
        v4f wb = *(const v4f*)(Wrow + kk + 4);
        v4f xa = *(const v4f*)(x + kk);
        v4f xb = *(const v4f*)(x + kk + 4);
        v2f a0 = {wa.x, wa.y}, a1 = {wa.z, wa.w};
        v2f a2 = {wb.x, wb.y}, a3 = {wb.z, wb.w};
        v2f b0 = {xa.x, xa.y}, b1 = {xa.z, xa.w};
        v2f b2 = {xb.x, xb.y}, b3 = {xb.z, xb.w};
        acc = __builtin_amdgcn_wmma_f32_16x16x4_f32(false, a0, false, b0, (short)0, acc, false, false);
        acc = __builtin_amdgcn_wmma_f32_16x16x4_f32(false, a1, false, b1, (short)0, acc, false, false);
        acc = __builtin_amdgcn_wmma_f32_16x16x4_f32(false, a2, false, b2, (short)0, acc, false, false);
        acc = __builtin_amdgcn_wmma_f32_16x16x4_f32(false, a3, false, b3, (short)0, acc, false, false);
    }
#else
    // Fallback: redundant per-lane FMA dot (only lanes 0 / 16 results are stored).
    for (int vi = 0; vi < 8; ++vi) {
        const float* __restrict__ r = W + (size_t)(rowBase + half * 8 + vi) * FAN;
        float s = 0.f;
        for (int k = k0; k < k0 + KCHUNK; ++k) s = fmaf(r[k], x[k], s);
        acc[vi] = s;
    }
#endif

    // Lane 0 holds rows 0..7 (col 0), lane 16 holds rows 8..15 (col 0).
    if ((lane & 15) == 0) {
        float* p = partial + (size_t)split * FAN + rowBase + half * 8;
        v4f lo = {acc[0], acc[1], acc[2], acc[3]};
        v4f hi = {acc[4], acc[5], acc[6], acc[7]};
        *(v4f*)(p)     = lo;
        *(v4f*)(p + 4) = hi;
    }
}

// ---------------------------------------------------------------------------
// Kernel B: reduce split-K partials, leaky integrate, sigmoid.
// out[0..FAN)      = outputs = sigmoid(v_new)
// out[FAN..2FAN)   = v_new
// ---------------------------------------------------------------------------
__global__ __launch_bounds__(256) void reduce_sigmoid_kernel(
    const float* __restrict__ pA, const float* __restrict__ pB,
    const float* __restrict__ v, float* __restrict__ out)
{
    const int o = blockIdx.x * blockDim.x + threadIdx.x;
    float s = 0.f;
#pragma unroll
    for (int sp = 0; sp < NSPLIT; ++sp)
        s += pA[sp * FAN + o] + pB[sp * FAN + o];
    const float vn = 0.1f * v[o] + s;          // (1 - LEAK) * v + ff + fb
    out[o]       = 1.f / (1.f + expf(-vn));    // sigmoid
    out[FAN + o] = vn;
}

// ---------------------------------------------------------------------------
// Threefry2x32 (20 rounds) — exact JAX PRNG core.
// ---------------------------------------------------------------------------
__device__ __forceinline__ unsigned rotl32(unsigned x, int r) {
    return (x << r) | (x >> (32 - r));
}

__device__ __forceinline__ void threefry2x32(unsigned k0, unsigned k1,
                                             unsigned x0, unsigned x1,
                                             unsigned& o0, unsigned& o1)
{
    const unsigned ks2 = k0 ^ k1 ^ 0x1BD11BDAu;
    x0 += k0; x1 += k1;
    x0 += x1; x1 = rotl32(x1, 13); x1 ^= x0;
    x0 += x1; x1 = rotl32(x1, 15); x1 ^= x0;
    x0 += x1; x1 = rotl32(x1, 26); x1 ^= x0;
    x0 += x1; x1 = rotl32(x1,  6); x1 ^= x0;
    x0 += k1; x1 += ks2 + 1u;
    x0 += x1; x1 = rotl32(x1, 17); x1 ^= x0;
    x0 += x1; x1 = rotl32(x1, 29); x1 ^= x0;
    x0 += x1; x1 = rotl32(x1, 16); x1 ^= x0;
    x0 += x1; x1 = rotl32(x1, 24); x1 ^= x0;
    x0 += ks2; x1 += k0 + 2u;
    x0 += x1; x1 = rotl32(x1, 13); x1 ^= x0;
    x0 += x1; x1 = rotl32(x1, 15); x1 ^= x0;
    x0 += x1; x1 = rotl32(x1, 26); x1 ^= x0;
    x0 += x1; x1 = rotl32(x1,  6); x1 ^= x0;
    x0 += k0; x1 += k1 + 3u;
    x0 += x1; x1 = rotl32(x1, 17); x1 ^= x0;
    x0 += x1; x1 = rotl32(x1, 29); x1 ^= x0;
    x0 += x1; x1 = rotl32(x1, 16); x1 ^= x0;
    x0 += x1; x1 = rotl32(x1, 24); x1 ^= x0;
    x0 += k1; x1 += ks2 + 4u;
    x0 += x1; x1 = rotl32(x1, 13); x1 ^= x0;
    x0 += x1; x1 = rotl32(x1, 15); x1 ^= x0;
    x0 += x1; x1 = rotl32(x1, 26); x1 ^= x0;
    x0 += x1; x1 = rotl32(x1,  6); x1 ^= x0;
    x0 += ks2; x1 += k0 + 5u;
    o0 = x0; o1 = x1;
}

__device__ __forceinline__ float bits_to_unit_float(unsigned b) {
    return __uint_as_float((b >> 9) | 0x3f800000u) - 1.f;
}

// ---------------------------------------------------------------------------
// Kernel C: spikify + STDP traces.
// key(42) -> (0, 42); split -> k1 = (tf(0,2).x0, tf(1,3).x0),
//                              k2 = (tf(0,2).x1, tf(1,3).x1)
// uniform(k, 8192): element i uses tf(k, (i&4095, 4096+(i&4095))),
//                   picks .x0 for i<4096 else .x1.
// out[2FAN..3FAN) = Apre_new ; out[3FAN..4FAN) = Apost_new
// ---------------------------------------------------------------------------
__global__ __launch_bounds__(256) void stdp_vectors_kernel(
    const float* __restrict__ inputs, const float* __restrict__ Apre,
    const float* __restrict__ Apost, const float* __restrict__ out_head, // outputs at d_out[0..FAN)
    float* __restrict__ ApreN, float* __restrict__ ApostN,
    float* __restrict__ in_spk, float* __restrict__ out_spk)
{
    const int i = blockIdx.x * blockDim.x + threadIdx.x;

    unsigned a0, b0, a1, b1;
    threefry2x32(0u, 42u, 0u, 2u, a0, b0);   // lane0 of split
    threefry2x32(0u, 42u, 1u, 3u, a1, b1);   // lane1 of split
    // subkey1 = (a0, a1), subkey2 = (b0, b1)

    const unsigned j = (unsigned)(i & 4095);
    unsigned r0, r1;

    threefry2x32(a0, a1, j, 4096u + j, r0, r1);
    const float u1 = bits_to_unit_float(i < 4096 ? r0 : r1);
    const float isp = (inputs[i] > u1) ? 1.f : 0.f;

    threefry2x32(b0, b1, j, 4096u + j, r0, r1);
    const float u2 = bits_to_unit_float(i < 4096 ? r0 : r1);
    const float osp = (out_head[i] > u2) ? 1.f : 0.f;

    const float apre  = Apre[i]  * 0.95f + isp;   // STDP_DECAY = 1 - 1/20
    const float apost = Apost[i] * 0.95f + osp;

    ApreN[i]  = apre;
    ApostN[i] = apost;
    in_spk[i]  = isp;
    out_spk[i] = osp;
}

// ---------------------------------------------------------------------------
// Kernel D: grad_new = grad_ff + outer(Apost_new, in_spk) - outer(out_spk, Apre_new)
// 512 MB of traffic — b128 loads/stores, one row per blockIdx.y.
// ---------------------------------------------------------------------------
__global__ __launch_bounds__(256) void grad_kernel(
    const float* __restrict__ grad_ff,
    const float* __restrict__ ApostN, const float* __restrict__ out_spk,
    const float* __restrict__ in_spk, const float* __restrict__ ApreN,
    float* __restrict__ grad_out)
{
    const int o  = blockIdx.y;
    const int c4 = (blockIdx.x * 256 + threadIdx.x) * 4;
    const size_t base = (size_t)o * FAN + c4;

    v4f g   = *(const v4f*)(grad_ff + base);
    v4f isv = *(const v4f*)(in_spk + c4);
    v4f apv = *(const v4f*)(ApreN + c4);
    const float apo = ApostN[o];
    const float osp = out_spk[o];

    g.x = fmaf(apo, isv.x, g.x) - osp * apv.x;
    g.y = fmaf(apo, isv.y, g.y) - osp * apv.y;
    g.z = fmaf(apo, isv.z, g.z) - osp * apv.z;
    g.w = fmaf(apo, isv.w, g.w) - osp * apv.w;

    *(v4f*)(grad_out + base) = g;
}

// ---------------------------------------------------------------------------
// Launch: inputs dict order = {inputs, c, v, W_ff, W_fb, Apre, Apost, grad_ff}
// d_out = [outputs | v_new | Apre_new | Apost_new | grad_new] (f32)
// ws: [pA: NSPLIT*FAN | pB: NSPLIT*FAN | in_spk: FAN | out_spk: FAN] f32
// ---------------------------------------------------------------------------
extern "C" void kernel_launch(void* const* d_in, const int* in_sizes, int n_in,
                              void* d_out, int out_size, void* d_ws, size_t ws_size,
                              hipStream_t stream)
{
    const float* inputs  = (const float*)d_in[0];
    const float* c       = (const float*)d_in[1];
    const float* v       = (const float*)d_in[2];
    const float* W_ff    = (const float*)d_in[3];
    const float* W_fb    = (const float*)d_in[4];
    const float* Apre    = (const float*)d_in[5];
    const float* Apost   = (const float*)d_in[6];
    const float* grad_ff = (const float*)d_in[7];

    float* out = (float*)d_out;
    float* ws  = (float*)d_ws;

    float* pA      = ws;
    float* pB      = ws + (size_t)NSPLIT * FAN;
    float* in_spk  = ws + (size_t)2 * NSPLIT * FAN;
    float* out_spk = in_spk + FAN;

    float* outputs   = out;                 // [0, FAN)
    float* apre_new  = out + 2 * FAN;       // [2FAN, 3FAN)
    float* apost_new = out + 3 * FAN;       // [3FAN, 4FAN)
    float* grad_new  = out + 4 * FAN;       // [4FAN, 4FAN + FAN*FAN)

    // A: split-K WMMA mat-vec partials (ff and fb)
    dim3 gA(FAN / 16 / 8, NSPLIT);          // (64, 16) blocks x 256 threads = 8192 waves
    matvec_partial_kernel<<<gA, 256, 0, stream>>>(W_ff, inputs, pA);
    matvec_partial_kernel<<<gA, 256, 0, stream>>>(W_fb, c, pB);

    // B: reduce + leaky integrate + sigmoid -> outputs, v_new
    reduce_sigmoid_kernel<<<FAN / 256, 256, 0, stream>>>(pA, pB, v, out);

    // C: threefry spikes + Apre/Apost traces
    stdp_vectors_kernel<<<FAN / 256, 256, 0, stream>>>(
        inputs, Apre, Apost, outputs, apre_new, apost_new, in_spk, out_spk);

    // D: rank-2 grad update (dominant memory stream)
    grad_kernel<<<dim3(FAN / 1024, FAN), 256, 0, stream>>>(
        grad_ff, apost_new, out_spk, in_spk, apre_new, grad_new);
}